// MultiHeadAttention_14963666059710
// MI455X (gfx1250) — compile-verified
//
#include <hip/hip_runtime.h>
#include <hip/hip_bf16.h>
#include <math.h>

// Problem constants (B=2, T=2048, C=1024, H=16, hd=64)
#define B_  2
#define T_  2048
#define C_  1024
#define H_  16
#define HD_ 64
#define N3_ 3072   // 3*C

typedef float v2f __attribute__((ext_vector_type(2)));
typedef float v8f __attribute__((ext_vector_type(8)));

// V_WMMA_F32_16X16X4_F32 : A 16x4 (2 VGPR), B 4x16 (2 VGPR), C/D 16x16 (8 VGPR)
__device__ __forceinline__ v8f wmma_f32x4(v2f a, v2f b, v8f c) {
  return __builtin_amdgcn_wmma_f32_16x16x4_f32(
      /*neg_a=*/false, a, /*neg_b=*/false, b,
      /*c_mod=*/(short)0, c, /*reuse_a=*/false, /*reuse_b=*/false);
}

// ---------------------------------------------------------------------------
// Kernel 1: qkv = x @ W_qkv   (M=B*T=4096, K=C=1024, N=3C=3072)
// One wave per 16x64 output tile; A operand reused for 4 N-subtiles.
// ---------------------------------------------------------------------------
__global__ __launch_bounds__(256) void qkv_gemm_kernel(
    const float* __restrict__ x, const float* __restrict__ w,
    float* __restrict__ qkv) {
  const int lane = threadIdx.x & 31;
  const int wid  = blockIdx.x * 8 + (threadIdx.x >> 5);
  const int tilesN = N3_ / 64;                 // 48
  const int tm = wid / tilesN;                 // 0..255
  const int tn = wid % tilesN;
  const int m    = lane & 15;
  const int half = lane >> 4;

  const float* xrow = x + (size_t)(tm * 16 + m) * C_;
  const int n0 = tn * 64;

  v8f acc0 = {}, acc1 = {}, acc2 = {}, acc3 = {};
  for (int k = 0; k < C_; k += 4) {
    const int kk = k + 2 * half;               // this lane's K pair
    float2 av = *reinterpret_cast<const float2*>(xrow + kk);
    v2f a; a.x = av.x; a.y = av.y;
    const float* wr0 = w + (size_t)kk * N3_ + n0 + m;
    const float* wr1 = wr0 + N3_;
    v2f b0; b0.x = wr0[0];  b0.y = wr1[0];
    v2f b1; b1.x = wr0[16]; b1.y = wr1[16];
    v2f b2; b2.x = wr0[32]; b2.y = wr1[32];
    v2f b3; b3.x = wr0[48]; b3.y = wr1[48];
    acc0 = wmma_f32x4(a, b0, acc0);
    acc1 = wmma_f32x4(a, b1, acc1);
    acc2 = wmma_f32x4(a, b2, acc2);
    acc3 = wmma_f32x4(a, b3, acc3);
  }
  for (int r = 0; r < 8; ++r) {                // C/D layout: (M=r+8*half, N=m)
    float* o = qkv + (size_t)(tm * 16 + r + 8 * half) * N3_ + n0 + m;
    o[0] = acc0[r]; o[16] = acc1[r]; o[32] = acc2[r]; o[48] = acc3[r];
  }
}

// ---------------------------------------------------------------------------
// Kernel 2: RoPE in-place on q (slot 0) and k (slot 1) inside qkv.
// Pair p of head-dim uses angle t * 10000^(-(p&15)/16)  (matches the
// reference's concat+[0::2] slicing).
// ---------------------------------------------------------------------------
__global__ __launch_bounds__(256) void rope_kernel(float* __restrict__ qkv) {
  int idx = blockIdx.x * 256 + threadIdx.x;    // B*2*T*H*32 threads
  const int p = idx & 31;          idx >>= 5;
  const int h = idx & 15;          idx >>= 4;
  const int t = idx & (T_ - 1);    idx >>= 11;
  const int s = idx & 1;           idx >>= 1;  // 0=q, 1=k
  const int b = idx;
  const int pm = p & 15;
  // ln(10000)/16 = 0.575646273248512
  const float angle = (float)t * expf(-0.575646273248512f * (float)pm);
  const float c  = cosf(angle);
  const float sn = sinf(angle);
  const size_t base =
      ((size_t)(b * T_ + t) * 3 + s) * C_ + h * HD_ + 2 * p;
  const float x1 = qkv[base];
  const float x2 = qkv[base + 1];
  qkv[base]     = x1 * c - x2 * sn;
  qkv[base + 1] = x2 * c + x1 * sn;
}

// butterfly reductions across the 16-lane half (matches WMMA C-layout rows)
__device__ __forceinline__ float halfmax16(float v) {
  for (int msk = 1; msk < 16; msk <<= 1) v = fmaxf(v, __shfl_xor(v, msk, 32));
  return v;
}
__device__ __forceinline__ float halfsum16(float v) {
  for (int msk = 1; msk < 16; msk <<= 1) v += __shfl_xor(v, msk, 32);
  return v;
}

// ---------------------------------------------------------------------------
// Kernel 3: fused attention. One wave per (b, h, 16-row q tile).
// Pass 1: streaming row max / sum-exp over all key tiles (flash style).
// Pass 2: recompute scores, write normalized attn once, O += P@V via WMMA
//         (P transposed C-layout -> A-layout through a 1KB per-wave LDS tile).
// ---------------------------------------------------------------------------
__global__ __launch_bounds__(256) void attn_kernel(
    const float* __restrict__ qkv, float* __restrict__ attn,
    float* __restrict__ outh) {
  __shared__ float ptile[8 * 256];             // 1KB per wave
  const int lane = threadIdx.x & 31;
  const int wib  = threadIdx.x >> 5;
  const int wid  = blockIdx.x * 8 + wib;
  const int qt = wid & 127;
  const int h  = (wid >> 7) & 15;
  const int b  = wid >> 11;
  const int m    = lane & 15;
  const int half = lane >> 4;
  const int q0 = qt * 16;
  float* lds = ptile + wib * 256;

  const float slope = exp2f(-0.5f * (float)(h + 1));   // ALiBi, H=16
  const float scale = 0.125f;                          // 1/sqrt(64)
  const float qposBase = (float)(q0 + 8 * half);

  // q tile -> registers (A layout: lane m holds row q0+m, K pair 2*half+{0,1})
  const float* qrow = qkv + ((size_t)(b * T_ + q0 + m) * 3 + 0) * C_ + h * HD_;
  v2f qa[16];
  for (int ks = 0; ks < 16; ++ks) {
    float2 v = *reinterpret_cast<const float2*>(qrow + ks * 4 + 2 * half);
    qa[ks].x = v.x; qa[ks].y = v.y;
  }

  float rm[8], rl[8];
  for (int r = 0; r < 8; ++r) { rm[r] = -3.0e38f; rl[r] = 0.0f; }

  // ---- pass 1: running softmax stats ----
  for (int kt = 0; kt < T_ / 16; ++kt) {
    const float* krow =
        qkv + ((size_t)(b * T_ + kt * 16 + m) * 3 + 1) * C_ + h * HD_;
    v8f acc = {};
    for (int ks = 0; ks < 16; ++ks) {
      float2 v = *reinterpret_cast<const float2*>(krow + ks * 4 + 2 * half);
      v2f kb; kb.x = v.x; kb.y = v.y;        // B layout: B[k][n=key]=K[key][k]
      acc = wmma_f32x4(qa[ks], kb, acc);
    }
    const float kpos = (float)(kt * 16 + m);
    for (int r = 0; r < 8; ++r) {
      float s = acc[r] * scale - slope * fabsf(kpos - (qposBase + (float)r));
      float tmax = halfmax16(s);
      float nm = fmaxf(rm[r], tmax);
      float psum = halfsum16(__expf(s - nm));
      rl[r] = rl[r] * __expf(rm[r] - nm) + psum;
      rm[r] = nm;
    }
  }

  // ---- pass 2: write attn + accumulate O = P @ V ----
  v8f o0 = {}, o1 = {}, o2 = {}, o3 = {};
  float* attnBase = attn + (size_t)(b * H_ + h) * T_ * T_;
  for (int kt = 0; kt < T_ / 16; ++kt) {
    const float* krow =
        qkv + ((size_t)(b * T_ + kt * 16 + m) * 3 + 1) * C_ + h * HD_;
    v8f acc = {};
    for (int ks = 0; ks < 16; ++ks) {
      float2 v = *reinterpret_cast<const float2*>(krow + ks * 4 + 2 * half);
      v2f kb; kb.x = v.x; kb.y = v.y;
      acc = wmma_f32x4(qa[ks], kb, acc);
    }
    const float kpos = (float)(kt * 16 + m);
    for (int r = 0; r < 8; ++r) {
      float s = acc[r] * scale - slope * fabsf(kpos - (qposBase + (float)r));
      float p = __expf(s - rm[r]) / rl[r];
      attnBase[(size_t)(q0 + r + 8 * half) * T_ + kt * 16 + m] = p;
      lds[(r + 8 * half) * 16 + m] = p;      // C-layout -> LDS row-major tile
    }
    // order the LDS stores before the transposed reload (same-wave DS is
    // in-order; this also stops compiler reordering via the memory clobber)
    asm volatile("s_wait_dscnt 0" ::: "memory");

    const size_t vbase = ((size_t)(b * T_ + kt * 16) * 3 + 2) * C_ + h * HD_;
    for (int kk = 0; kk < 4; ++kk) {
      float2 pv =
          *reinterpret_cast<const float2*>(lds + m * 16 + kk * 4 + 2 * half);
      v2f pa; pa.x = pv.x; pa.y = pv.y;      // A layout of P
      const int key = kk * 4 + 2 * half;
      const float* v0 = qkv + vbase + (size_t)key * N3_;
      const float* v1 = v0 + N3_;
      v2f vb0; vb0.x = v0[m];      vb0.y = v1[m];
      v2f vb1; vb1.x = v0[16 + m]; vb1.y = v1[16 + m];
      v2f vb2; vb2.x = v0[32 + m]; vb2.y = v1[32 + m];
      v2f vb3; vb3.x = v0[48 + m]; vb3.y = v1[48 + m];
      o0 = wmma_f32x4(pa, vb0, o0);
      o1 = wmma_f32x4(pa, vb1, o1);
      o2 = wmma_f32x4(pa, vb2, o2);
      o3 = wmma_f32x4(pa, vb3, o3);
    }
  }

  for (int r = 0; r < 8; ++r) {
    float* orow =
        outh + (size_t)(b * T_ + q0 + r + 8 * half) * C_ + h * HD_ + m;
    orow[0] = o0[r]; orow[16] = o1[r]; orow[32] = o2[r]; orow[48] = o3[r];
  }
}

// ---------------------------------------------------------------------------
// Kernel 4: out = outh @ W_out + b_out   (M=4096, K=1024, N=1024)
// ---------------------------------------------------------------------------
__global__ __launch_bounds__(256) void out_gemm_kernel(
    const float* __restrict__ a, const float* __restrict__ w,
    const float* __restrict__ bias, float* __restrict__ out) {
  const int lane = threadIdx.x & 31;
  const int wid  = blockIdx.x * 8 + (threadIdx.x >> 5);
  const int tilesN = C_ / 64;                  // 16
  const int tm = wid / tilesN;
  const int tn = wid % tilesN;
  const int m    = lane & 15;
  const int half = lane >> 4;

  const float* arow = a + (size_t)(tm * 16 + m) * C_;
  const int n0 = tn * 64;

  v8f acc0 = {}, acc1 = {}, acc2 = {}, acc3 = {};
  for (int k = 0; k < C_; k += 4) {
    const int kk = k + 2 * half;
    float2 av = *reinterpret_cast<const float2*>(arow + kk);
    v2f aa; aa.x = av.x; aa.y = av.y;
    const float* wr0 = w + (size_t)kk * C_ + n0 + m;
    const float* wr1 = wr0 + C_;
    v2f b0; b0.x = wr0[0];  b0.y = wr1[0];
    v2f b1; b1.x = wr0[16]; b1.y = wr1[16];
    v2f b2; b2.x = wr0[32]; b2.y = wr1[32];
    v2f b3; b3.x = wr0[48]; b3.y = wr1[48];
    acc0 = wmma_f32x4(aa, b0, acc0);
    acc1 = wmma_f32x4(aa, b1, acc1);
    acc2 = wmma_f32x4(aa, b2, acc2);
    acc3 = wmma_f32x4(aa, b3, acc3);
  }
  for (int r = 0; r < 8; ++r) {
    float* o = out + (size_t)(tm * 16 + r + 8 * half) * C_ + n0 + m;
    o[0]  = acc0[r] + bias[n0 + m];
    o[16] = acc1[r] + bias[n0 + 16 + m];
    o[32] = acc2[r] + bias[n0 + 32 + m];
    o[48] = acc3[r] + bias[n0 + 48 + m];
  }
}

// ---------------------------------------------------------------------------
extern "C" void kernel_launch(void* const* d_in, const int* in_sizes, int n_in,
                              void* d_out, int out_size, void* d_ws,
                              size_t ws_size, hipStream_t stream) {
  const float* x     = (const float*)d_in[0];
  const float* W_qkv = (const float*)d_in[1];
  const float* W_out = (const float*)d_in[2];
  const float* b_out = (const float*)d_in[3];

  float* out  = (float*)d_out;                          // B*T*C
  float* attn = out + (size_t)B_ * T_ * C_;             // B*H*T*T
  float* qkv  = (float*)d_ws;                           // B*T*3C  (48 MB)
  float* outh = qkv + (size_t)B_ * T_ * N3_;            // B*T*C   (16 MB)

  // 1) qkv projection: 256 M-tiles * 48 N-tiles = 12288 waves
  qkv_gemm_kernel<<<12288 / 8, 256, 0, stream>>>(x, W_qkv, qkv);
  // 2) RoPE on q and k slots: B*2*T*H*32 = 4,194,304 threads
  rope_kernel<<<(B_ * 2 * T_ * H_ * 32) / 256, 256, 0, stream>>>(qkv);
  // 3) fused attention: B*H*(T/16) = 4096 waves
  attn_kernel<<<4096 / 8, 256, 0, stream>>>(qkv, attn, outh);
  // 4) output projection: 256 M-tiles * 16 N-tiles = 4096 waves
  out_gemm_kernel<<<4096 / 8, 256, 0, stream>>>(outh, W_out, b_out, out);
}